// DownBlock_472446403331
// MI455X (gfx1250) — compile-verified
//
#include <hip/hip_runtime.h>
#include <math.h>

// Problem constants (match reference)
#define NN   4096
#define EE   131072
#define CC   256
#define TT   256
#define KK   2048   // POOL_K

typedef __attribute__((ext_vector_type(2))) float v2f;
typedef __attribute__((ext_vector_type(8))) float v8f;

// ---------------------------------------------------------------------------
// CDNA5 async global -> LDS copy (ASYNCcnt tracked, no VGPR staging).
// VDST operand = LDS byte offset (low 32 bits of the flat shared address:
// the LDS aperture occupies addr[63:32], so addr[31:0] is the LDS offset).
// ---------------------------------------------------------------------------
__device__ __forceinline__ void async_ld_b128(const float* lds_dst, const float* gsrc) {
    const unsigned lds_off = (unsigned)(uintptr_t)lds_dst;
    asm volatile("global_load_async_to_lds_b128 %0, %1, off"
                 :: "v"(lds_off), "v"(gsrc) : "memory");
}
__device__ __forceinline__ void wait_async0() {
    asm volatile("s_wait_asynccnt 0x0" ::: "memory");
}

// ---------------------------------------------------------------------------
// Pooling score: score[i] = tanh( dot(x[i,:], w) / ||w|| )
// ---------------------------------------------------------------------------
__global__ __launch_bounds__(256) void score_kernel(const float* __restrict__ x,
                                                    const float* __restrict__ pw,
                                                    float* __restrict__ score) {
    __shared__ float red[256];
    const int row = blockIdx.x;
    const int t = threadIdx.x;
    const float w = pw[t];
    const float xv = x[(size_t)row * CC + t];
    red[t] = xv * w;
    __syncthreads();
    for (int s = 128; s > 0; s >>= 1) { if (t < s) red[t] += red[t + s]; __syncthreads(); }
    const float dot = red[0];
    __syncthreads();
    red[t] = w * w;
    __syncthreads();
    for (int s = 128; s > 0; s >>= 1) { if (t < s) red[t] += red[t + s]; __syncthreads(); }
    if (t == 0) score[row] = tanhf(dot / sqrtf(red[0]));
}

// ---------------------------------------------------------------------------
// Full bitonic sort of 4096 (score,idx) pairs, descending score / ascending idx.
// ---------------------------------------------------------------------------
__global__ __launch_bounds__(1024) void topk_sort_kernel(const float* __restrict__ score,
                                                         int* __restrict__ perm,
                                                         float* __restrict__ sel,
                                                         int* __restrict__ invp) {
    __shared__ float skey[NN];
    __shared__ int   sidx[NN];
    const int t = threadIdx.x;
    for (int i = t; i < NN; i += 1024) { skey[i] = score[i]; sidx[i] = i; }
    __syncthreads();
    for (int k = 2; k <= NN; k <<= 1) {
        for (int j = k >> 1; j > 0; j >>= 1) {
            for (int i = t; i < NN; i += 1024) {
                const int ixj = i ^ j;
                if (ixj > i) {
                    const float k1 = skey[i], k2 = skey[ixj];
                    const int   i1 = sidx[i], i2 = sidx[ixj];
                    const bool firstWins = (k1 > k2) || (k1 == k2 && i1 < i2);
                    const bool up = ((i & k) == 0);
                    const bool doSwap = up ? !firstWins : firstWins;
                    if (doSwap) { skey[i] = k2; skey[ixj] = k1; sidx[i] = i2; sidx[ixj] = i1; }
                }
            }
            __syncthreads();
        }
    }
    for (int i = t; i < NN; i += 1024) {
        const int oi = sidx[i];
        invp[oi] = (i < KK) ? i : -1;
        if (i < KK) { perm[i] = oi; sel[i] = skey[i]; }
    }
}

// ---------------------------------------------------------------------------
// Scatter edges directly into Ar = A[perm,:] and Ac = A[:,perm]
// ---------------------------------------------------------------------------
__global__ __launch_bounds__(256) void edge_scatter_kernel(const int* __restrict__ ei,
                                                           const float* __restrict__ ew,
                                                           const int* __restrict__ invp,
                                                           float* __restrict__ Ar,
                                                           float* __restrict__ Ac) {
    const int e = blockIdx.x * 256 + threadIdx.x;
    if (e >= EE) return;
    const int r = ei[e];
    const int c = ei[EE + e];
    const float w = ew[e];
    const int ir = invp[r];
    if (ir >= 0) atomicAdd(&Ar[(size_t)ir * NN + c], w);
    const int ic = invp[c];
    if (ic >= 0) atomicAdd(&Ac[(size_t)r * KK + ic], w);
}

// Self loops: A[i,i] = 1.0 (set, overwriting accumulated weight)
__global__ __launch_bounds__(256) void diag_set_kernel(const int* __restrict__ invp,
                                                       float* __restrict__ Ar,
                                                       float* __restrict__ Ac) {
    const int i = blockIdx.x * 256 + threadIdx.x;
    if (i >= NN) return;
    const int ir = invp[i];
    if (ir >= 0) { Ar[(size_t)ir * NN + i] = 1.0f; Ac[(size_t)i * KK + ir] = 1.0f; }
}

// ---------------------------------------------------------------------------
// FP32 WMMA GEMM with async-LDS double buffering.
// C[M,N] = A[M,K] @ B[K,N].  M%64==0, N%128==0, K%32==0.
// 256 threads = 8 wave32 waves; block tile 64x128; each wave computes a
// 32x32 patch as 2x2 v_wmma_f32_16x16x4_f32 accumulators.
// ---------------------------------------------------------------------------
#define BM 64
#define BN 128
#define BK 32
#define LDA 36   // padded row stride (dwords): 36*m mod 64 distinct for m in [0,16)

__global__ __launch_bounds__(256) void gemm_wmma_f32(const float* __restrict__ A,
                                                     const float* __restrict__ B,
                                                     float* __restrict__ C,
                                                     int M, int N, int K) {
    __shared__ __align__(16) float As[2][BM * LDA];   // 2 x 9216 B
    __shared__ __align__(16) float Bs[2][BK * BN];    // 2 x 16384 B
    const int tid  = threadIdx.x;
    const int lane = tid & 31;
    const int wave = tid >> 5;
    const int wm = wave & 1;     // wave grid 2 (M) x 4 (N)
    const int wn = wave >> 1;
    const int bm = blockIdx.y * BM;
    const int bn = blockIdx.x * BN;

    v8f acc[2][2] = {};

    const int lh    = lane & 15;
    const int khalf = (lane >> 4) << 1;  // 0 or 2 (A/B frag K-half per lane group)
    const int hi8   = (lane >> 4) << 3;  // 0 or 8 (C/D row offset per lane group)

    // Async staging of one K-tile into buffer `buf`
    auto stage = [&](int k0, int buf) {
        int f = tid;
        #pragma unroll
        for (int it = 0; it < 2; ++it, f += 256) {      // A: 64x32 = 512 float4s
            const int r  = f >> 3;
            const int c4 = (f & 7) << 2;
            async_ld_b128(&As[buf][r * LDA + c4],
                          &A[(size_t)(bm + r) * K + k0 + c4]);
        }
        f = tid;
        #pragma unroll
        for (int it = 0; it < 4; ++it, f += 256) {      // B: 32x128 = 1024 float4s
            const int r  = f >> 5;
            const int c4 = (f & 31) << 2;
            async_ld_b128(&Bs[buf][r * BN + c4],
                          &B[(size_t)(k0 + r) * N + bn + c4]);
        }
    };

    stage(0, 0);
    wait_async0();
    __syncthreads();

    const int ntiles = K / BK;
    for (int t = 0; t < ntiles; ++t) {
        const int buf = t & 1;
        if (t + 1 < ntiles) stage((t + 1) * BK, buf ^ 1);

        const float* as = As[buf];
        const float* bs = Bs[buf];
        #pragma unroll
        for (int kk = 0; kk < BK; kk += 4) {
            v2f a[2], b[2];
            #pragma unroll
            for (int mi = 0; mi < 2; ++mi) {
                const int r = wm * 32 + mi * 16 + lh;
                const float* p = &as[r * LDA + kk + khalf];
                a[mi].x = p[0];
                a[mi].y = p[1];
            }
            #pragma unroll
            for (int ni = 0; ni < 2; ++ni) {
                const int c = wn * 32 + ni * 16 + lh;
                b[ni].x = bs[(kk + khalf) * BN + c];
                b[ni].y = bs[(kk + khalf + 1) * BN + c];
            }
            #pragma unroll
            for (int mi = 0; mi < 2; ++mi)
                #pragma unroll
                for (int ni = 0; ni < 2; ++ni)
                    acc[mi][ni] = __builtin_amdgcn_wmma_f32_16x16x4_f32(
                        false, a[mi], false, b[ni], (short)0, acc[mi][ni], false, false);
        }
        wait_async0();      // own prefetch landed in LDS
        __syncthreads();    // all waves done reading buf; safe to overwrite next iter
    }

    #pragma unroll
    for (int mi = 0; mi < 2; ++mi) {
        #pragma unroll
        for (int ni = 0; ni < 2; ++ni) {
            const int rbase = bm + wm * 32 + mi * 16 + hi8;
            const int cidx  = bn + wn * 32 + ni * 16 + lh;
            #pragma unroll
            for (int v = 0; v < 8; ++v)
                C[(size_t)(rbase + v) * N + cidx] = acc[mi][ni][v];
        }
    }
}

// ---------------------------------------------------------------------------
// Zero diag of A_p, compute deg = rowsum(A_hat) with +2 self loop, dinv.
// ---------------------------------------------------------------------------
__global__ __launch_bounds__(256) void deg_kernel(float* __restrict__ Ap,
                                                  float* __restrict__ dinv) {
    __shared__ float red[256];
    const int i = blockIdx.x;
    const int t = threadIdx.x;
    float s = 0.f;
    for (int j = t; j < KK; j += 256) s += Ap[(size_t)i * KK + j];
    red[t] = s;
    __syncthreads();
    for (int st = 128; st > 0; st >>= 1) { if (t < st) red[t] += red[t + st]; __syncthreads(); }
    if (t == 0) {
        const float diag = Ap[(size_t)i * KK + i];
        const float deg = red[0] - diag + 2.0f;
        Ap[(size_t)i * KK + i] = 0.0f;
        dinv[i] = deg > 0.f ? rsqrtf(deg) : 0.f;
    }
}

__global__ __launch_bounds__(256) void an_kernel(const float* __restrict__ Ap,
                                                 const float* __restrict__ dinv,
                                                 float* __restrict__ An) {
    const size_t idx = (size_t)blockIdx.x * 256 + threadIdx.x;
    const int i = (int)(idx >> 11);
    const int j = (int)(idx & (KK - 1));
    const float v = Ap[idx] + (i == j ? 2.0f : 0.0f);
    An[idx] = dinv[i] * v * dinv[j];
}

__global__ __launch_bounds__(256) void xp_kernel(const float* __restrict__ x,
                                                 const int* __restrict__ perm,
                                                 const float* __restrict__ sel,
                                                 float* __restrict__ xp) {
    const size_t idx = (size_t)blockIdx.x * 256 + threadIdx.x;
    const int i = (int)(idx >> 8);
    const int c = (int)(idx & (CC - 1));
    xp[idx] = x[(size_t)perm[i] * CC + c] * sel[i];
}

__global__ __launch_bounds__(256) void temb_kernel(const float* __restrict__ t,
                                                   const float* __restrict__ Wt,
                                                   const float* __restrict__ bt,
                                                   float* __restrict__ temb) {
    const int c = threadIdx.x;
    float s = bt[c];
    for (int k = 0; k < TT; ++k) s += t[k] * Wt[k * CC + c];
    temb[c] = fmaxf(s, 0.f);
}

// relu(H + bias) -> LayerNorm(g, be) [+ temb]
__global__ __launch_bounds__(256) void ln_kernel(const float* __restrict__ H,
                                                 const float* __restrict__ bias,
                                                 const float* __restrict__ g,
                                                 const float* __restrict__ be,
                                                 const float* __restrict__ temb,
                                                 int addTemb,
                                                 float* __restrict__ out) {
    __shared__ float red[256];
    const int i = blockIdx.x;
    const int t = threadIdx.x;
    const float v = fmaxf(H[(size_t)i * CC + t] + bias[t], 0.f);
    red[t] = v;
    __syncthreads();
    for (int s = 128; s > 0; s >>= 1) { if (t < s) red[t] += red[t + s]; __syncthreads(); }
    const float mu = red[0] * (1.f / CC);
    __syncthreads();
    const float d = v - mu;
    red[t] = d * d;
    __syncthreads();
    for (int s = 128; s > 0; s >>= 1) { if (t < s) red[t] += red[t + s]; __syncthreads(); }
    const float var = red[0] * (1.f / CC);
    float y = d * rsqrtf(var + 1e-5f) * g[t] + be[t];
    if (addTemb) y += temb[t];
    out[(size_t)i * CC + t] = y;
}

__global__ __launch_bounds__(256) void permout_kernel(const int* __restrict__ perm,
                                                      float* __restrict__ out) {
    const int i = blockIdx.x * 256 + threadIdx.x;
    if (i < KK) out[i] = (float)perm[i];
}

// ---------------------------------------------------------------------------
extern "C" void kernel_launch(void* const* d_in, const int* in_sizes, int n_in,
                              void* d_out, int out_size, void* d_ws, size_t ws_size,
                              hipStream_t stream) {
    const float* x   = (const float*)d_in[0];
    const int*   ei  = (const int*)d_in[1];
    const float* ew  = (const float*)d_in[2];
    const float* t   = (const float*)d_in[3];
    const float* pw  = (const float*)d_in[4];
    const float* W1  = (const float*)d_in[5];
    const float* b1  = (const float*)d_in[6];
    const float* g1  = (const float*)d_in[7];
    const float* be1 = (const float*)d_in[8];
    const float* W2  = (const float*)d_in[9];
    const float* b2  = (const float*)d_in[10];
    const float* g2  = (const float*)d_in[11];
    const float* be2 = (const float*)d_in[12];
    const float* Wt  = (const float*)d_in[13];
    const float* bt  = (const float*)d_in[14];

    char* ws = (char*)d_ws;
    size_t off = 0;
    float* Ar   = (float*)(ws + off); off += (size_t)KK * NN * 4;   // 32 MB
    float* Ac   = (float*)(ws + off); off += (size_t)NN * KK * 4;   // 32 MB
    float* An   = (float*)(ws + off); off += (size_t)KK * KK * 4;   // 16 MB
    float* XW   = (float*)(ws + off); off += (size_t)KK * CC * 4;
    float* Hbuf = (float*)(ws + off); off += (size_t)KK * CC * 4;
    float* h1   = (float*)(ws + off); off += (size_t)KK * CC * 4;
    float* xp   = (float*)(ws + off); off += (size_t)KK * CC * 4;
    float* score= (float*)(ws + off); off += (size_t)NN * 4;
    float* sel  = (float*)(ws + off); off += (size_t)KK * 4;
    int*   perm = (int*)(ws + off);   off += (size_t)KK * 4;
    int*   invp = (int*)(ws + off);   off += (size_t)NN * 4;
    float* temb = (float*)(ws + off); off += (size_t)CC * 4;
    float* dinv = (float*)(ws + off); off += (size_t)KK * 4;

    float* hOut    = (float*)d_out;
    float* ApOut   = hOut + (size_t)KK * CC;
    float* permOut = ApOut + (size_t)KK * KK;

    // Zero the scatter targets every call (graph replay safe)
    hipMemsetAsync(Ar, 0, (size_t)KK * NN * 4, stream);
    hipMemsetAsync(Ac, 0, (size_t)NN * KK * 4, stream);

    score_kernel<<<NN, 256, 0, stream>>>(x, pw, score);
    topk_sort_kernel<<<1, 1024, 0, stream>>>(score, perm, sel, invp);
    edge_scatter_kernel<<<EE / 256, 256, 0, stream>>>(ei, ew, invp, Ar, Ac);
    diag_set_kernel<<<NN / 256, 256, 0, stream>>>(invp, Ar, Ac);

    // A_p = Ar @ Ac  (2048 x 2048, K = 4096), written to output slot
    gemm_wmma_f32<<<dim3(KK / BN, KK / BM), 256, 0, stream>>>(Ar, Ac, ApOut, KK, KK, NN);
    deg_kernel<<<KK, 256, 0, stream>>>(ApOut, dinv);
    an_kernel<<<(KK * KK) / 256, 256, 0, stream>>>(ApOut, dinv, An);

    xp_kernel<<<(KK * CC) / 256, 256, 0, stream>>>(x, perm, sel, xp);
    temb_kernel<<<1, 256, 0, stream>>>(t, Wt, bt, temb);

    // conv1: H = A_n @ (x_p @ W1); relu+bias+LN1 (+temb) fused in ln_kernel
    gemm_wmma_f32<<<dim3(CC / BN, KK / BM), 256, 0, stream>>>(xp, W1, XW, KK, CC, CC);
    gemm_wmma_f32<<<dim3(CC / BN, KK / BM), 256, 0, stream>>>(An, XW, Hbuf, KK, CC, KK);
    ln_kernel<<<KK, 256, 0, stream>>>(Hbuf, b1, g1, be1, temb, 1, h1);

    // conv2
    gemm_wmma_f32<<<dim3(CC / BN, KK / BM), 256, 0, stream>>>(h1, W2, XW, KK, CC, CC);
    gemm_wmma_f32<<<dim3(CC / BN, KK / BM), 256, 0, stream>>>(An, XW, Hbuf, KK, CC, KK);
    ln_kernel<<<KK, 256, 0, stream>>>(Hbuf, b2, g2, be2, temb, 0, hOut);

    permout_kernel<<<KK / 256, 256, 0, stream>>>(perm, permOut);
}